// triplet_loss_77214922048069
// MI455X (gfx1250) — compile-verified
//
#include <hip/hip_runtime.h>
#include <hip/hip_bf16.h>

// Problem constants (reference: N=8192, D=128, C=512, ALPHA=0.2)
#define NROW 8192
#define DIM  128
#define NLAB 512
#define ALPHA 0.2f
#define LDS_PITCH 132   // 132 % 64 = 4 -> strided B reads hit distinct banks

typedef __attribute__((ext_vector_type(2))) float v2f;
typedef __attribute__((ext_vector_type(8))) float v8f;

// ---------------- prepass kernels ----------------

__global__ void k_init(float* out, int* minidx) {
    int t = blockIdx.x * blockDim.x + threadIdx.x;
    if (t == 0) out[0] = 0.0f;
    if (t < NLAB) minidx[t] = 0x7fffffff;
}

__global__ void k_minidx(const int* __restrict__ lab, int* __restrict__ minidx) {
    int i = blockIdx.x * blockDim.x + threadIdx.x;
    if (i < NROW) atomicMin(&minidx[lab[i]], i);
}

__global__ void k_sq(const float* __restrict__ X, float* __restrict__ sq) {
    int i = blockIdx.x * blockDim.x + threadIdx.x;
    if (i >= NROW) return;
    const float4* row = (const float4*)(X + i * DIM);
    float s = 0.0f;
    #pragma unroll
    for (int k = 0; k < DIM / 4; ++k) {
        float4 v = row[k];
        s += v.x * v.x + v.y * v.y + v.z * v.z + v.w * v.w;
    }
    sq[i] = s;
}

// anchor id, ap[i] = ||x_i - x_{a(i)}||^2, pos flag
__global__ void k_ap(const float* __restrict__ X, const int* __restrict__ lab,
                     const int* __restrict__ minidx, const float* __restrict__ sq,
                     int* __restrict__ anchor, float* __restrict__ ap, int* __restrict__ pos) {
    int i = blockIdx.x * blockDim.x + threadIdx.x;
    if (i >= NROW) return;
    int a = minidx[lab[i]];
    anchor[i] = a;
    pos[i] = (a != i) ? 1 : 0;
    const float4* xi = (const float4*)(X + i * DIM);
    const float4* xa = (const float4*)(X + a * DIM);
    float dot = 0.0f;
    #pragma unroll
    for (int k = 0; k < DIM / 4; ++k) {
        float4 u = xi[k], v = xa[k];
        dot += u.x * v.x + u.y * v.y + u.z * v.z + u.w * v.w;
    }
    ap[i] = sq[i] + sq[a] - 2.0f * dot;
}

// ---------------- main WMMA kernel ----------------
// 64 blocks x 256 threads (8 wave32). Wave w owns i-tile rows [i0, i0+16).
// Double-buffered LDS j-tiles: global loads for tile j+1 are issued before
// the 64-WMMA compute block of tile j so the L2 round trip overlaps compute;
// one barrier per iteration.

__global__ void __launch_bounds__(256, 1)
k_triplet(const float* __restrict__ X, const int* __restrict__ lab,
          const int* __restrict__ anchor, const float* __restrict__ sq,
          const float* __restrict__ ap, const int* __restrict__ pos,
          float* __restrict__ out) {
    __shared__ float bt[2][16 * LDS_PITCH];
    __shared__ float red[8];

    const int tid  = threadIdx.x;
    const int lane = tid & 31;
    const int wave = tid >> 5;
    const int half = lane >> 4;   // 0: lanes 0-15, 1: lanes 16-31
    const int l15  = lane & 15;
    const int i0   = blockIdx.x * 128 + wave * 16;

    // ----- A-layout register tiles: rows of X and gathered anchor rows -----
    // f32 A 16x4 layout: lane&15 = M; lanes 0-15 hold K={0,1}, lanes 16-31 K={2,3}
    const int arow  = i0 + l15;
    const int krow  = anchor[arow];
    const int kbase = 2 * half;
    v2f ax[32], aa[32];
    #pragma unroll
    for (int kk = 0; kk < 32; ++kk) {
        ax[kk] = *(const v2f*)(X + arow * DIM + 4 * kk + kbase);
        aa[kk] = *(const v2f*)(X + krow * DIM + 4 * kk + kbase);
    }

    // ----- per-lane epilogue constants for its 8 C/D elements -----
    // C/D layout: VGPR v, lanes 0-15 -> M=v, lanes 16-31 -> M=v+8; N = lane&15
    float e1[8], e2[8];  // ap - sq_a, ap - sq_i (pos-mask folded in)
    int   labi[8];
    #pragma unroll
    for (int v = 0; v < 8; ++v) {
        int r = i0 + 8 * half + v;
        float apr = ap[r];
        int   ar  = anchor[r];
        labi[v] = lab[r];
        if (pos[r]) {
            e1[v] = apr - sq[ar];
            e2[v] = apr - sq[r];
        } else {
            e1[v] = -1.0e30f;   // hinge window can never open
            e2[v] = -1.0e30f;
        }
    }

    // per-thread mapping for cooperative tile staging (16 rows x 32 float4)
    const int str = tid >> 5;     // rows 0..7 (second chunk: +8)
    const int stc = tid & 31;     // float4 column 0..31

    // ----- prologue: stage j-tile 0 into buffer 0 -----
    float4 s0, s1;
    {
        const float4* src = (const float4*)X;   // j0 = 0, row stride = 32 float4
        s0 = src[tid];
        s1 = src[tid + 256];
    }
    *(float4*)&bt[0][ str      * LDS_PITCH + stc * 4] = s0;
    *(float4*)&bt[0][(str + 8) * LDS_PITCH + stc * 4] = s1;
    __syncthreads();

    float acc = 0.0f;

    for (int jt = 0; jt < NROW / 16; ++jt) {
        const int j0 = jt * 16;
        const float* buf = bt[jt & 1];
        const bool haveNext = (jt + 1) < (NROW / 16);

        // issue global loads for the NEXT tile early (overlap with WMMA)
        if (haveNext) {
            const float4* src = (const float4*)(X + (j0 + 16) * DIM);
            s0 = src[tid];
            s1 = src[tid + 256];
        }

        const int   jc   = j0 + l15;
        const int   labj = lab[jc];
        const float sqj  = sq[jc];

        v8f accX = {};  // x_i . x_j
        v8f accA = {};  // x_{a(i)} . x_j
        #pragma unroll
        for (int kk = 0; kk < 32; ++kk) {
            // B 4x16 tile: lane&15 = N column, K = kbase..kbase+1 (mirrors A)
            v2f b = *(const v2f*)&buf[l15 * LDS_PITCH + 4 * kk + kbase];
            accX = __builtin_amdgcn_wmma_f32_16x16x4_f32(
                false, ax[kk], false, b, (short)0, accX, false, false);
            accA = __builtin_amdgcn_wmma_f32_16x16x4_f32(
                false, aa[kk], false, b, (short)0, accA, false, false);
        }

        // fused hinge epilogue: D = sq_i + sq_j - 2G; d = ap - D
        #pragma unroll
        for (int v = 0; v < 8; ++v) {
            if (labi[v] != labj) {
                float d1 = e1[v] - sqj + 2.0f * accA[v];
                float d2 = e2[v] - sqj + 2.0f * accX[v];
                float t1 = (d1 < 0.0f && d1 + ALPHA > 0.0f) ? d1 + ALPHA : 0.0f;
                float t2 = (d2 < 0.0f && d2 + ALPHA > 0.0f) ? d2 + ALPHA : 0.0f;
                acc += t1 + t2;
            }
        }

        // stash the staged next tile into the other buffer
        if (haveNext) {
            float* nbuf = bt[(jt + 1) & 1];
            *(float4*)&nbuf[ str      * LDS_PITCH + stc * 4] = s0;
            *(float4*)&nbuf[(str + 8) * LDS_PITCH + stc * 4] = s1;
        }
        // single barrier: iteration jt's reads of buf[jt&1] complete here,
        // so iteration jt+1 may safely overwrite buf[jt&1]
        __syncthreads();
    }

    // ----- reduction: wave shuffle -> LDS -> one float atomic per block -----
    #pragma unroll
    for (int off = 16; off > 0; off >>= 1)
        acc += __shfl_down(acc, off, 32);
    if (lane == 0) red[wave] = acc;
    __syncthreads();
    if (tid == 0) {
        float s = 0.0f;
        #pragma unroll
        for (int w = 0; w < 8; ++w) s += red[w];
        unsafeAtomicAdd(out, s);
    }
}

// ---------------- launch ----------------

extern "C" void kernel_launch(void* const* d_in, const int* in_sizes, int n_in,
                              void* d_out, int out_size, void* d_ws, size_t ws_size,
                              hipStream_t stream) {
    const float* X   = (const float*)d_in[0];
    const int*   lab = (const int*)d_in[1];
    float*       out = (float*)d_out;

    char* ws = (char*)d_ws;
    int*   minidx = (int*)(ws);                       // 512 ints
    int*   anchor = (int*)(ws + 4096);                // 8192 ints
    float* sq     = (float*)(ws + 4096 + 32768);      // 8192 f32
    float* ap     = (float*)(ws + 4096 + 65536);      // 8192 f32
    int*   pos    = (int*)(ws + 4096 + 98304);        // 8192 ints

    k_init  <<<1, 512, 0, stream>>>(out, minidx);
    k_minidx<<<NROW / 256, 256, 0, stream>>>(lab, minidx);
    k_sq    <<<NROW / 256, 256, 0, stream>>>(X, sq);
    k_ap    <<<NROW / 256, 256, 0, stream>>>(X, lab, minidx, sq, anchor, ap, pos);
    k_triplet<<<NROW / 128, 256, 0, stream>>>(X, lab, anchor, sq, ap, pos, out);
}